// HabanaOptimizerSparseSgd_74904229642566
// MI455X (gfx1250) — compile-verified
//
#include <hip/hip_runtime.h>
#include <hip/hip_bf16.h>

// ---------------------------------------------------------------------------
// Sparse SGD table update for MI455X (gfx1250).
//   out[0 : V*D]        = weights  - scatter_add(lr * grads over indices)
//   out[V*D : 2*V*D]    = moments  (pass-through)
// Bulk table copy (1 GB) goes through the Tensor Data Mover (TDM):
//   global -> LDS -> global, tracked with TENSORcnt, no VGPR traffic.
// Scatter uses native f32 global atomics (duplicate indices accumulate).
// Roofline: ~2.2 GB moved => ~95 us at 23.3 TB/s; no matrix structure, so
// the CDNA5-specific win is the async TDM path + native global atomics.
// ---------------------------------------------------------------------------

#define CHUNK_BYTES  65536u                 // 64 KB per workgroup (fits LDS)
#define CHUNK_ELEMS8 (CHUNK_BYTES / 8u)     // 8192 8-byte elements per tile

typedef __attribute__((ext_vector_type(4))) unsigned int v4u;
typedef __attribute__((ext_vector_type(8))) int          v8i;
typedef __attribute__((ext_vector_type(4))) int          v4i;

// Build D# group0: count=1 (valid), lds_addr, global_addr[56:0], type=2.
__device__ __forceinline__ v4u tdm_group0(unsigned long long gaddr,
                                          unsigned int lds_addr) {
  v4u g0;
  g0.x = 1u;                                                  // count=1, no gather
  g0.y = lds_addr;                                            // LDS byte address
  g0.z = (unsigned int)(gaddr & 0xFFFFFFFFull);               // global_addr[31:0]
  g0.w = (unsigned int)((gaddr >> 32) & 0x01FFFFFFull)        // global_addr[56:32]
       | (2u << 30);                                          // type = 2 ("image")
  return g0;
}

// One wave per 64 KB chunk: TDM load tile into LDS, then TDM store to dest.
__global__ __launch_bounds__(32) void tdm_bulk_copy(
    const float* __restrict__ weights,
    const float* __restrict__ moments,
    float* __restrict__ out,
    unsigned int chunks_per_half) {
  __shared__ __attribute__((aligned(128))) unsigned char lds_buf[CHUNK_BYTES];

  unsigned int chunk = blockIdx.x;
  unsigned long long src;
  if (chunk < chunks_per_half) {
    src = (unsigned long long)(uintptr_t)weights
        + (unsigned long long)chunk * CHUNK_BYTES;
  } else {
    src = (unsigned long long)(uintptr_t)moments
        + (unsigned long long)(chunk - chunks_per_half) * CHUNK_BYTES;
  }
  unsigned long long dst = (unsigned long long)(uintptr_t)out
                         + (unsigned long long)chunk * CHUNK_BYTES;
  unsigned int lds_addr = (unsigned int)(uintptr_t)lds_buf;

  // D# group1 (shared by load & store): data_size=3 (8B elements),
  // tile = 8192 x 1, tensor_dim0 = 8192 (stride 8192), no padding/iterate.
  v8i g1;
  g1[0] = (int)(3u << 16);                               // data_size=8B; mask=0
  g1[1] = (int)((CHUNK_ELEMS8 & 0xFFFFu) << 16);         // tensor_dim0[15:0]
  g1[2] = (int)((CHUNK_ELEMS8 >> 16) | (1u << 16));      // tensor_dim0[31:16] | tensor_dim1=1
  g1[3] = (int)((CHUNK_ELEMS8 & 0xFFFFu) << 16);         // tile_dim0 = 8192
  g1[4] = 1;                                             // tile_dim1=1, tile_dim2=0
  g1[5] = (int)CHUNK_ELEMS8;                             // tensor_dim0_stride
  g1[6] = 0;
  g1[7] = 0;
  v4i gz4 = {0, 0, 0, 0};                                // groups 2/3 unused (2D)
  v8i gz8 = {0, 0, 0, 0, 0, 0, 0, 0};                    // extra group (6-arg form)

  // DMA: memory -> LDS
  __builtin_amdgcn_tensor_load_to_lds(tdm_group0(src, lds_addr),
                                      g1, gz4, gz4, gz8, 0);
  __builtin_amdgcn_s_wait_tensorcnt(0);
  // DMA: LDS -> memory
  __builtin_amdgcn_tensor_store_from_lds(tdm_group0(dst, lds_addr),
                                         g1, gz4, gz4, gz8, 0);
  __builtin_amdgcn_s_wait_tensorcnt(0);
}

// One wave32 per gradient row (D=128): lane handles 4 consecutive floats.
// weights_out[idx[row]] += (-lr) * grad[row], via native f32 global atomics.
__global__ __launch_bounds__(256) void sparse_sgd_scatter(
    const float* __restrict__ grads,
    const int*   __restrict__ indices,
    const float* __restrict__ lr_ptr,
    const int*   __restrict__ vc_ptr,
    float*       __restrict__ out_weights,
    int n_rows) {
  int row  = (int)(blockIdx.x * 8u + (threadIdx.x >> 5));
  int lane = (int)(threadIdx.x & 31u);
  if (row >= n_rows) return;

  int valid_count = vc_ptr[0];            // uniform scalar load
  if (row >= valid_count) return;         // masked / padded rows

  float neg_lr = -lr_ptr[0];
  unsigned long long idx = (unsigned long long)(unsigned int)indices[row];

  // 128-bit vector load of this lane's 4 gradient values (global_load_b128)
  const float4 g =
      ((const float4*)(grads + (unsigned long long)row * 128ull))[lane];

  float* dst = out_weights + idx * 128ull + (unsigned int)lane * 4u;
  unsafeAtomicAdd(dst + 0, neg_lr * g.x);   // global_atomic_add_f32
  unsafeAtomicAdd(dst + 1, neg_lr * g.y);
  unsafeAtomicAdd(dst + 2, neg_lr * g.z);
  unsafeAtomicAdd(dst + 3, neg_lr * g.w);
}

extern "C" void kernel_launch(void* const* d_in, const int* in_sizes, int n_in,
                              void* d_out, int out_size, void* d_ws, size_t ws_size,
                              hipStream_t stream) {
  // setup_inputs() order: gradients, weights, moments, indices, learning_rate, valid_count
  const float* grads   = (const float*)d_in[0];
  const float* weights = (const float*)d_in[1];
  const float* moments = (const float*)d_in[2];
  const int*   indices = (const int*)d_in[3];
  const float* lr      = (const float*)d_in[4];
  const int*   vc      = (const int*)d_in[5];
  float*       out     = (float*)d_out;

  // Bulk copy: weights -> out[0:VD], moments -> out[VD:2VD].
  // V*D*4 bytes = 512 MB per half, exactly divisible by 64 KB chunks.
  unsigned long long half_bytes = (unsigned long long)in_sizes[1] * 4ull;
  unsigned int chunks_per_half  = (unsigned int)(half_bytes / CHUNK_BYTES);
  unsigned int total_chunks     = chunks_per_half * 2u;
  tdm_bulk_copy<<<total_chunks, 32, 0, stream>>>(weights, moments, out,
                                                 chunks_per_half);

  // Scatter-add on top of the copied weights (same stream -> ordered).
  int n_rows = in_sizes[0] / 128;                 // N gradient rows (D=128)
  int blocks = (n_rows + 7) / 8;                  // 8 waves per block
  sparse_sgd_scatter<<<blocks, 256, 0, stream>>>(grads, indices, lr, vc,
                                                 out, n_rows);
}